// DiscreteMMSE_62294205661211
// MI455X (gfx1250) — compile-verified
//
#include <hip/hip_runtime.h>
#include <hip/hip_bf16.h>
#include <stdint.h>

typedef __attribute__((ext_vector_type(2))) float v2f;
typedef __attribute__((ext_vector_type(8))) float v8f;

#define B_ 32
#define P_ 256
#define D_ 64
#define T_ 4096
#define TSLAB 128
#define NSLAB (T_ / TSLAB)        // 32
#define NTHREADS 128
#define NWAVES (NTHREADS / 32)    // 4 waves (wave32)
#define WPAD 68                   // padded row stride (floats) for W/A tiles
#define PPAD 132                  // padded row stride (floats) for pred tile
#define ROWB (WPAD * 4)           // padded row stride in bytes (272)

// Online-softmax merge of (m, Z, A) triples:
//   Z = sum_t exp(S_t - m),  A = sum_t exp(S_t - m) * pred_t
__device__ __forceinline__ void lse_merge(float& m, float& z, float& a,
                                          float m2, float z2, float a2) {
  float M  = fmaxf(m, m2);
  float e1 = __expf(m - M);
  float e2 = __expf(m2 - M);
  z = z * e1 + z2 * e2;
  a = a * e1 + a2 * e2;
  m = M;
}

// CDNA5 async copy: 16 bytes global -> LDS, tracked by ASYNCcnt (GVS mode).
__device__ __forceinline__ void async_cp16(uint32_t lds_addr, const void* sbase,
                                           uint32_t voff) {
  asm volatile("global_load_async_to_lds_b128 %0, %1, %2"
               :: "v"(lds_addr), "v"(voff), "s"(sbase)
               : "memory");
}

__device__ __forceinline__ void wait_async0() {
  asm volatile("s_wait_asynccnt 0x0" ::: "memory");
}

// Phase 1: one workgroup per (task-slab, batch).
// pred via V_WMMA_F32_16X16X4_F32; A tiles double-buffered with
// GLOBAL_LOAD_ASYNC_TO_LDS_B128 so staging overlaps WMMA + softmax reduction.
extern "C" __global__ __launch_bounds__(NTHREADS)
void mmse_phase1(const float* __restrict__ data,      // (B, P, D)
                 const float* __restrict__ targets,   // (B, P)
                 const float* __restrict__ taskpool,  // (T, D, 1): W[d,t]=taskpool[t*D+d]
                 float* __restrict__ partials) {      // (B, P, NSLAB, 3)
  extern __shared__ float smem[];
  float* Wlds  = smem;                          // TSLAB * WPAD  (W_lds[t][d])
  float* Alds0 = Wlds + TSLAB * WPAD;           // 16 * WPAD     (A double buffer)
  float* Alds1 = Alds0 + 16 * WPAD;             // 16 * WPAD
  float* Plds  = Alds1 + 16 * WPAD;             // 16 * PPAD     (pred[p][t])
  float* Tlds  = Plds + 16 * PPAD;              // P_
  float* Red   = Tlds + P_;                     // 16 * NWAVES * 3

  const int tid  = threadIdx.x;
  const int lane = tid & 31;
  const int wave = tid >> 5;
  const int b    = blockIdx.y;
  const int slab = blockIdx.x;
  const int t0   = slab * TSLAB;

  // LDS byte offsets for async destinations (generic ptr low 32 bits = LDS offset)
  const uint32_t WldsB  = (uint32_t)(uintptr_t)Wlds;
  const uint32_t AldsB0 = (uint32_t)(uintptr_t)Alds0;
  const uint32_t AldsB1 = (uint32_t)(uintptr_t)Alds1;

  const float* wbase = taskpool + (size_t)t0 * D_;   // W slab base (row-major, 256B rows)
  const float* dbase = data + (size_t)b * P_ * D_;   // this batch's data

  // ---- Prologue: async-stage W slab (padded rows) + A tile 0; targets normally ----
  for (int i = tid; i < TSLAB * 16; i += NTHREADS) { // 16B chunks: row t, chunk c
    int t = i >> 4, c = i & 15;
    async_cp16(WldsB + t * ROWB + c * 16, wbase, (uint32_t)(t * 256 + c * 16));
  }
  for (int i = tid; i < 16 * 16; i += NTHREADS) {    // A tile 0
    int r = i >> 4, c = i & 15;
    async_cp16(AldsB0 + r * ROWB + c * 16, dbase, (uint32_t)(r * 256 + c * 16));
  }
  for (int i = tid; i < P_; i += NTHREADS) Tlds[i] = targets[b * P_ + i];

  const int h   = lane >> 4;           // lane half selects K pair (ISA 16x4 f32 layout)
  const int l15 = lane & 15;
  const int n0a = wave * 32;           // this wave's two 16-col tiles (local task)
  const int n0b = wave * 32 + 16;

  float cum = 0.0f;                    // exclusive prefix of scores for task `tid`

  for (int pi = 0; pi < P_ / 16; ++pi) {
    const int cur = pi & 1;
    wait_async0();                     // current A buffer (and W on pi==0) landed
    __syncthreads();                   // visible to all waves; Plds/Red reusable

    if (pi + 1 < P_ / 16) {            // prefetch next A tile into the other buffer
      uint32_t dstB = cur ? AldsB0 : AldsB1;
      for (int i = tid; i < 16 * 16; i += NTHREADS) {
        int r = i >> 4, c = i & 15;
        async_cp16(dstB + r * ROWB + c * 16, dbase,
                   (uint32_t)(((pi + 1) * 16 + r) * 256 + c * 16));
      }
    }

    // ---- pred tile: two 16x16 WMMA accumulators per wave, K=64 in 16 steps ----
    const float* Acur = cur ? Alds1 : Alds0;
    v8f c0 = {}; v8f c1 = {};
    const float* arow  = Acur + l15 * WPAD;          // A: M = lane&15
    const float* brow0 = Wlds + (n0a + l15) * WPAD;  // B: N = lane&15 (W^T in LDS)
    const float* brow1 = Wlds + (n0b + l15) * WPAD;
#pragma unroll
    for (int j = 0; j < 16; ++j) {
      int k = 4 * j + 2 * h;                         // lanes 0-15: K=(0,1); 16-31: K=(2,3)
      v2f a  = *(const v2f*)(arow  + k);
      v2f b0 = *(const v2f*)(brow0 + k);
      v2f b1 = *(const v2f*)(brow1 + k);
      c0 = __builtin_amdgcn_wmma_f32_16x16x4_f32(false, a, false, b0, (short)0, c0, false, false);
      c1 = __builtin_amdgcn_wmma_f32_16x16x4_f32(false, a, false, b1, (short)0, c1, false, false);
    }
    // D layout: VGPR r -> M=r (lanes 0-15), M=r+8 (lanes 16-31); N = lane&15
#pragma unroll
    for (int r = 0; r < 8; ++r) {
      int row = r + 8 * h;
      Plds[row * PPAD + n0a + l15] = c0[r];
      Plds[row * PPAD + n0b + l15] = c1[r];
    }
    __syncthreads();

    // ---- Per-row slab-softmax partial: max-reduce, 1 exp/thread, sum-reduce ----
#pragma unroll 1
    for (int r = 0; r < 16; ++r) {
      float pred  = Plds[r * PPAD + tid];            // task `tid`, row p = pi*16+r
      float err   = Tlds[pi * 16 + r] - pred;
      float score = -0.5f * err * err;               // log_norm cancels in softmax
      float m = cum;
#pragma unroll
      for (int off = 16; off > 0; off >>= 1)
        m = fmaxf(m, __shfl_xor(m, off, 32));        // wave max of prefix S
      float e = __expf(cum - m);
      float z = e, a = e * pred;
#pragma unroll
      for (int off = 16; off > 0; off >>= 1) {       // butterfly sums
        z += __shfl_xor(z, off, 32);
        a += __shfl_xor(a, off, 32);
      }
      if (lane == 0) {
        float* q = Red + (r * NWAVES + wave) * 3;
        q[0] = m; q[1] = z; q[2] = a;
      }
      cum += score;                                  // exclusive -> inclusive after use
    }
    __syncthreads();

    if (tid < 16) {                                  // combine the 4 wave partials
      float* q = Red + tid * NWAVES * 3;
      float m = q[0], z = q[1], a = q[2];
      for (int w = 1; w < NWAVES; ++w)
        lse_merge(m, z, a, q[w * 3 + 0], q[w * 3 + 1], q[w * 3 + 2]);
      int p = pi * 16 + tid;
      float* o = partials + (((size_t)b * P_ + p) * NSLAB + slab) * 3;
      o[0] = m; o[1] = z; o[2] = a;
    }
    // next iteration's top barrier covers Red/Plds reuse
  }
}

// Phase 2: merge the NSLAB partials per (b,p); out = A/Z.
extern "C" __global__ __launch_bounds__(256)
void mmse_phase2(const float* __restrict__ partials, float* __restrict__ out) {
  int i = blockIdx.x * 256 + threadIdx.x;            // i = b*P + p
  if (i >= B_ * P_) return;
  const float* q = partials + (size_t)i * NSLAB * 3;
  float m = q[0], z = q[1], a = q[2];
  for (int s = 1; s < NSLAB; ++s)
    lse_merge(m, z, a, q[s * 3 + 0], q[s * 3 + 1], q[s * 3 + 2]);
  out[i] = a / z;
}

extern "C" void kernel_launch(void* const* d_in, const int* in_sizes, int n_in,
                              void* d_out, int out_size, void* d_ws, size_t ws_size,
                              hipStream_t stream) {
  const float* data     = (const float*)d_in[0];   // (32, 256, 64)
  const float* targets  = (const float*)d_in[1];   // (32, 256)
  const float* taskpool = (const float*)d_in[2];   // (4096, 64, 1)
  float* out      = (float*)d_out;                 // (32, 256)
  float* partials = (float*)d_ws;                  // B*P*NSLAB*3 floats = 3 MB

  const size_t shmem = (size_t)(TSLAB * WPAD + 2 * 16 * WPAD + 16 * PPAD + P_ +
                                16 * NWAVES * 3) * sizeof(float);  // ~53.8 KB

  mmse_phase1<<<dim3(NSLAB, B_), NTHREADS, shmem, stream>>>(data, targets, taskpool, partials);
  mmse_phase2<<<dim3((B_ * P_ + 255) / 256), 256, 0, stream>>>(partials, out);
}